// MultiHeadSelfAttention_39694087750009
// MI455X (gfx1250) — compile-verified
//
#include <hip/hip_runtime.h>

typedef float v2f __attribute__((ext_vector_type(2)));
typedef float v8f __attribute__((ext_vector_type(8)));
typedef int   b128_t __attribute__((vector_size(16)));   // matches builtin's expected pointee

#if __has_builtin(__builtin_amdgcn_global_load_async_to_lds_b128)
#define ASYNC_LDS 1
#else
#define ASYNC_LDS 0
#endif

#define BM 128
#define BN 64
#define BK 32
#define LD 36             // row stride (floats): 16B-aligned rows, conflict-free b64 frag reads
#define ASZ (BM * LD)     // 4608 floats per A buffer
#define BSZ (BN * LD)     // 2304 floats per B buffer

// 16-byte global -> LDS copy; async (ASYNCcnt) on CDNA5, VGPR round-trip otherwise.
__device__ __forceinline__ void cp16(const float* g, float* l) {
#if ASYNC_LDS
    __builtin_amdgcn_global_load_async_to_lds_b128(
        (__attribute__((address_space(1))) b128_t*)(void*)g,
        (__attribute__((address_space(3))) b128_t*)(void*)l, 0, 0);
#else
    *(float4*)l = *(const float4*)g;
#endif
}

__device__ __forceinline__ void async_wait0() {
#if ASYNC_LDS
#if __has_builtin(__builtin_amdgcn_s_wait_asynccnt)
    __builtin_amdgcn_s_wait_asynccnt(0);
#else
    asm volatile("s_wait_asynccnt 0x0" ::: "memory");
#endif
#endif
}

#define WMMA_F32(a, b, c) \
    __builtin_amdgcn_wmma_f32_16x16x4_f32(false, (a), false, (b), (short)0, (c), false, false)

// ---------------------------------------------------------------------------
// Y[m,n] = sum_k X[m,k] * W[n,k] (+ bias[n])   i.e.  Y = X @ W^T + b
// Block tile 128x64 (8 waves, 32x32 per wave), fp32 WMMA 16x16x4,
// double-buffered LDS fed by async-to-LDS copies.
// Both LDS tiles are natural row-major [row][k] with stride 36:
//   A frag: v2f at As[(wm+lr)*36 + kk + 2*hh]          (lane half owns K+2)
//   B frag: v2f at Bs[(wn+lr)*36 + kk + 2*hh]          (VGPR v, half h -> K = v + 2h)
// ---------------------------------------------------------------------------
__global__ __launch_bounds__(256)
void mhsa_gemm_xwt(const float* __restrict__ X, const float* __restrict__ W,
                   const float* __restrict__ bias, float* __restrict__ Y,
                   int M, int N, int K)
{
    __shared__ float As[2 * ASZ];
    __shared__ float Bs[2 * BSZ];

    const int tid  = threadIdx.x;
    const int lane = tid & 31;
    const int lr   = lane & 15;
    const int hh   = lane >> 4;
    const int wid  = tid >> 5;

    const int bn = blockIdx.x * BN;
    const int bm = blockIdx.y * BM;
    const int wm = (wid & 3) * 32;
    const int wn = (wid >> 2) * 32;

    // staging map: 8 threads x 16B cover one 32-float row
    const int srow = tid >> 3;
    const int sc4  = (tid & 7) * 4;

    v8f acc00 = {}; v8f acc01 = {}; v8f acc10 = {}; v8f acc11 = {};

    const int T = K / BK;

    // ---- prologue: stage tile 0 into buffer 0 ----
    #pragma unroll
    for (int u = 0; u < 4; ++u)
        cp16(X + (size_t)(bm + srow + u * 32) * K + sc4,
             As + (srow + u * 32) * LD + sc4);
    #pragma unroll
    for (int u = 0; u < 2; ++u)
        cp16(W + (size_t)(bn + srow + u * 32) * K + sc4,
             Bs + (srow + u * 32) * LD + sc4);
    async_wait0();
    __syncthreads();

    for (int t = 0; t < T; ++t) {
        const float* Ab = As + (t & 1) * ASZ;
        const float* Bb = Bs + (t & 1) * BSZ;

        // ---- prefetch tile t+1 into the other buffer (overlaps WMMAs below) ----
        if (t + 1 < T) {
            const int k0 = (t + 1) * BK;
            float* Ad = As + ((t + 1) & 1) * ASZ;
            float* Bd = Bs + ((t + 1) & 1) * BSZ;
            #pragma unroll
            for (int u = 0; u < 4; ++u)
                cp16(X + (size_t)(bm + srow + u * 32) * K + k0 + sc4,
                     Ad + (srow + u * 32) * LD + sc4);
            #pragma unroll
            for (int u = 0; u < 2; ++u)
                cp16(W + (size_t)(bn + srow + u * 32) * K + k0 + sc4,
                     Bd + (srow + u * 32) * LD + sc4);
        }

        // ---- fragment-pipelined 8 x (K=4) WMMA steps ----
        const float* Ap0 = Ab + (wm + lr) * LD + 2 * hh;
        const float* Ap1 = Ab + (wm + 16 + lr) * LD + 2 * hh;
        const float* Bp0 = Bb + (wn + lr) * LD + 2 * hh;
        const float* Bp1 = Bb + (wn + 16 + lr) * LD + 2 * hh;

        v2f a0 = *(const v2f*)(Ap0);
        v2f a1 = *(const v2f*)(Ap1);
        v2f b0 = *(const v2f*)(Bp0);
        v2f b1 = *(const v2f*)(Bp1);
        #pragma unroll
        for (int kk = 0; kk < BK; kk += 4) {
            v2f na0 = a0, na1 = a1, nb0 = b0, nb1 = b1;
            if (kk + 4 < BK) {
                na0 = *(const v2f*)(Ap0 + kk + 4);
                na1 = *(const v2f*)(Ap1 + kk + 4);
                nb0 = *(const v2f*)(Bp0 + kk + 4);
                nb1 = *(const v2f*)(Bp1 + kk + 4);
            }
            acc00 = WMMA_F32(a0, b0, acc00);
            acc01 = WMMA_F32(a0, b1, acc01);
            acc10 = WMMA_F32(a1, b0, acc10);
            acc11 = WMMA_F32(a1, b1, acc11);
            a0 = na0; a1 = na1; b0 = nb0; b1 = nb1;
        }

        async_wait0();       // tile t+1 fully resident
        __syncthreads();     // everyone done reading buffer t before it is reused
    }

    // ---- epilogue: C/D layout VGPR v -> M = v + 8*half, N = lane&15 ----
    const int n0 = bn + wn + lr;
    const int n1 = n0 + 16;
    const float bv0 = bias ? bias[n0] : 0.0f;
    const float bv1 = bias ? bias[n1] : 0.0f;
    #pragma unroll
    for (int v = 0; v < 8; ++v) {
        int m0 = bm + wm + v + 8 * hh;
        int m1 = m0 + 16;
        Y[(size_t)m0 * N + n0] = acc00[v] + bv0;
        Y[(size_t)m0 * N + n1] = acc01[v] + bv1;
        Y[(size_t)m1 * N + n0] = acc10[v] + bv0;
        Y[(size_t)m1 * N + n1] = acc11[v] + bv1;
    }
}

// ---------------------------------------------------------------------------
// Per-token head-gram attention (one wave per token, negligible FLOPs).
// att[i][j] = sum_d Q[i,d]*K[j,d]; mask -> -1e20; softmax(att/sqrt(768)) over j;
// O[i,d] = sum_j att[i,j]*V[j,d], stored [B][H][S][D] so the output-projection
// GEMM consumes the reference's raw [b,h,s,d]->[b,s,hd] reshape for free.
// ---------------------------------------------------------------------------
__global__ __launch_bounds__(256)
void mhsa_attn(const float* __restrict__ Q, const float* __restrict__ Kp,
               const float* __restrict__ Vp, const int* __restrict__ mask,
               float* __restrict__ O, int S)
{
    __shared__ float sm[8 * 1600];          // per wave: q[768] k[768] a[64]
    const int lane = threadIdx.x & 31;
    const int wid  = threadIdx.x >> 5;
    const int t    = blockIdx.x * 8 + wid;  // token index in [0, B*S)

    float* q = &sm[wid * 1600];
    float* k = q + 768;
    float* a = k + 768;

    const size_t rb = (size_t)t * 768;
    #pragma unroll
    for (int j = 0; j < 6; ++j) {           // 192 float4 per row / 32 lanes
        int idx = (lane + j * 32) * 4;
        *(float4*)&q[idx] = *(const float4*)&Q[rb + idx];
        *(float4*)&k[idx] = *(const float4*)&Kp[rb + idx];
    }
    const int mv = mask[t];

    // V rows in registers: lane owns dims {lane, lane+32, lane+64}
    float vr[8][3];
    #pragma unroll
    for (int j = 0; j < 8; ++j) {
        vr[j][0] = Vp[rb + j * 96 + lane];
        vr[j][1] = Vp[rb + j * 96 + lane + 32];
        vr[j][2] = Vp[rb + j * 96 + lane + 64];
    }

    // gram: each lane computes 2 of the 64 (i,j) entries
    {
        const int i  = lane >> 2;
        const int j0 = (lane & 3) * 2;
        float g0 = 0.f, g1 = 0.f;
        const float* qi  = &q[i * 96];
        const float* kj0 = &k[j0 * 96];
        const float* kj1 = kj0 + 96;
        for (int d = 0; d < 96; ++d) {
            float qv = qi[d];
            g0 += qv * kj0[d];
            g1 += qv * kj1[d];
        }
        a[i * 8 + j0]     = g0;
        a[i * 8 + j0 + 1] = g1;
    }

    // softmax over j (lanes 0..7, one row each); wave-internal DS ordering
    const float inv_scale = 0.03608439182435161f;  // 1/sqrt(768)
    if (lane < 8) {
        float r[8];
        float mx = -3.402823e38f;
        #pragma unroll
        for (int j = 0; j < 8; ++j) {
            float x = (mv == 0) ? -1e20f : a[lane * 8 + j];
            x *= inv_scale;
            r[j] = x;
            mx = fmaxf(mx, x);
        }
        float ssum = 0.f;
        #pragma unroll
        for (int j = 0; j < 8; ++j) { r[j] = __expf(r[j] - mx); ssum += r[j]; }
        float inv = 1.f / ssum;
        #pragma unroll
        for (int j = 0; j < 8; ++j) a[lane * 8 + j] = r[j] * inv;
    }

    // O = A @ V, stored [B][8][S][96]
    const int b = t / S;
    const int s = t - b * S;
    #pragma unroll
    for (int i = 0; i < 8; ++i) {
        float o0 = 0.f, o1 = 0.f, o2 = 0.f;
        #pragma unroll
        for (int j = 0; j < 8; ++j) {
            float aij = a[i * 8 + j];
            o0 += aij * vr[j][0];
            o1 += aij * vr[j][1];
            o2 += aij * vr[j][2];
        }
        size_t ob = ((size_t)(b * 8 + i) * S + s) * 96;
        O[ob + lane]      = o0;
        O[ob + lane + 32] = o1;
        O[ob + lane + 64] = o2;
    }
}

extern "C" void kernel_launch(void* const* d_in, const int* in_sizes, int n_in,
                              void* d_out, int out_size, void* d_ws, size_t ws_size,
                              hipStream_t stream) {
    (void)in_sizes; (void)n_in; (void)out_size; (void)ws_size;
    const float* values = (const float*)d_in[0];
    const float* keys   = (const float*)d_in[1];
    const float* query  = (const float*)d_in[2];
    const int*   mask   = (const int*)d_in[3];
    const float* Wv     = (const float*)d_in[4];
    const float* Wk     = (const float*)d_in[5];
    const float* Wq     = (const float*)d_in[6];
    const float* Wo     = (const float*)d_in[7];
    const float* bo     = (const float*)d_in[8];
    float* out = (float*)d_out;

    const int Bb = 4, S = 4096, E = 768;
    const int M = Bb * S;                                // 16384
    const size_t bytes = (size_t)M * E * sizeof(float);  // 48 MB

    char* ws = (char*)d_ws;
    float* Pv = (float*)(ws + 0 * bytes);
    float* Pk = (float*)(ws + 1 * bytes);
    float* Pq = (float*)(ws + 2 * bytes);
    float* AO = (float*)(ws + 3 * bytes);

    dim3 grid(E / BN, M / BM);                           // 12 x 128
    dim3 blk(256);
    mhsa_gemm_xwt<<<grid, blk, 0, stream>>>(values, Wv, nullptr, Pv, M, E, E);
    mhsa_gemm_xwt<<<grid, blk, 0, stream>>>(keys,   Wk, nullptr, Pk, M, E, E);
    mhsa_gemm_xwt<<<grid, blk, 0, stream>>>(query,  Wq, nullptr, Pq, M, E, E);
    mhsa_attn<<<M / 8, 256, 0, stream>>>(Pq, Pk, Pv, mask, AO, S);
    mhsa_gemm_xwt<<<grid, blk, 0, stream>>>(AO, Wo, bo, out, M, E, E);
}